// DisjointSet_83210696393434
// MI455X (gfx1250) — compile-verified
//
#include <hip/hip_runtime.h>

// DisjointSet densify + gather for MI455X (gfx1250).
//
// reference:  f_conv   = pointer-jump(father) to fixed point (== root id per node)
//             gathered = values[f_conv]
// outputs:    d_out = [ f_conv : int64[N] | gathered : float32[N] ]  (flat concat)
//
// Strategy: per-thread root chase (the fixed point of f=f[f] is the tree root,
// so no multi-pass global sync is needed). father(128MB i64) + values(64MB f32)
// = 192 MB fits exactly in MI455X's 192 MB L2, so the data-dependent chase
// loads are L2 hits. Outputs (192 MB) are streamed with TH=NT stores so the
// write traffic doesn't evict the tables from L2.
//
// MLP: 4 elements per thread, all 4 chases fused into one loop that issues 4
// independent b64 loads per iteration (single wait), quadrupling outstanding
// loads per wave vs. serialized chases. Converged elements re-load their own
// root line (L0 hit) which keeps the loop branch-free.

typedef __attribute__((ext_vector_type(2))) long long v2i64;
typedef __attribute__((ext_vector_type(4))) float     v4f32;

__global__ __launch_bounds__(256)
void ds_densify_gather(const long long* __restrict__ father,
                       const float*     __restrict__ values,
                       long long*       __restrict__ out_root,
                       float*           __restrict__ out_val,
                       long long        n)
{
  long long i0 = (((long long)blockIdx.x * (long long)blockDim.x) +
                  (long long)threadIdx.x) * 4;
  if (i0 >= n) return;

  if (i0 + 3 < n) {
    // Coalesced streaming reads: 2x global_load_b128.
    v2i64 fa = *reinterpret_cast<const v2i64*>(father + i0);
    v2i64 fb = *reinterpret_cast<const v2i64*>(father + i0 + 2);

    long long x0 = fa.x, x1 = fa.y, x2 = fb.x, x3 = fb.y;
    long long p0 = father[x0];
    long long p1 = father[x1];
    long long p2 = father[x2];
    long long p3 = father[x3];

    // Fused 4-way chase: 4 independent dependent-loads in flight per iteration.
    // For converged elements p==x, so x=p; p=f[x] is a fixed point (L0 hit).
    while ((p0 != x0) | (p1 != x1) | (p2 != x2) | (p3 != x3)) {
      x0 = p0; x1 = p1; x2 = p2; x3 = p3;
      p0 = father[x0];
      p1 = father[x1];
      p2 = father[x2];
      p3 = father[x3];
    }

    // gfx1250 global_prefetch_b8: warm the 4 values[] lines while the two NT
    // root stores are being issued below.
    __builtin_prefetch((const void*)(values + x0), 0, 0);
    __builtin_prefetch((const void*)(values + x1), 0, 0);
    __builtin_prefetch((const void*)(values + x2), 0, 0);
    __builtin_prefetch((const void*)(values + x3), 0, 0);

    // Streaming outputs: non-temporal (TH=NT) so L2 stays reserved for the
    // father/values tables.
    v2i64 ra; ra.x = x0; ra.y = x1;
    v2i64 rb; rb.x = x2; rb.y = x3;
    __builtin_nontemporal_store(ra, reinterpret_cast<v2i64*>(out_root + i0));
    __builtin_nontemporal_store(rb, reinterpret_cast<v2i64*>(out_root + i0 + 2));

    v4f32 vv;
    vv.x = values[x0];
    vv.y = values[x1];
    vv.z = values[x2];
    vv.w = values[x3];
    __builtin_nontemporal_store(vv, reinterpret_cast<v4f32*>(out_val + i0));
  } else {
    // Scalar tail (not taken for N = 2^24, kept for generality).
    for (long long i = i0; i < n && i < i0 + 4; ++i) {
      long long x = father[i];
      long long p = father[x];
      while (p != x) { x = p; p = father[x]; }
      __builtin_nontemporal_store(x, out_root + i);
      float v = values[x];
      __builtin_nontemporal_store(v, out_val + i);
    }
  }
}

extern "C" void kernel_launch(void* const* d_in, const int* in_sizes, int n_in,
                              void* d_out, int out_size, void* d_ws, size_t ws_size,
                              hipStream_t stream) {
  (void)n_in; (void)out_size; (void)d_ws; (void)ws_size;

  // Reference dtypes: father is int64 (128 MB table), values is float32.
  const long long* father = (const long long*)d_in[0];
  const float*     values = (const float*)d_in[1];
  const long long  n      = (long long)in_sizes[0];

  // Tuple output, concatenated flat: int64 roots first, then float32 gather.
  long long* out_root = (long long*)d_out;
  float*     out_val  = (float*)(out_root + n);

  const int  block    = 256;                 // 8 wave32 per block
  long long  nthreads = (n + 3) / 4;         // 4 elements per thread
  long long  grid     = (nthreads + block - 1) / block;

  ds_densify_gather<<<(unsigned int)grid, block, 0, stream>>>(
      father, values, out_root, out_val, n);
}